// PreconditionerNet_8753143349638
// MI455X (gfx1250) — compile-verified
//
#include <hip/hip_runtime.h>
#include <hip/hip_bf16.h>
#include <math.h>

// CDNA5 / gfx1250, wave32. Fully-fused sparse-conv net, fp32 WMMA path.
// Triangular-output specialization: upper tiles are pure zero-fill (no loads,
// no WMMA), lower tiles skip the triangular/softplus epilogue, only the 64
// diagonal tiles run the full epilogue.
typedef __attribute__((ext_vector_type(2))) float v2f;
typedef __attribute__((ext_vector_type(8))) float v8f;

__global__ __launch_bounds__(32)
void precond_net_kernel(const float* __restrict__ X, const int* __restrict__ MASK,
                        const float* __restrict__ W1, const float* __restrict__ B1,
                        const float* __restrict__ A1,
                        const float* __restrict__ W2, const float* __restrict__ B2,
                        const float* __restrict__ A2,
                        const float* __restrict__ W3, const float* __restrict__ B3,
                        const float* __restrict__ A3,
                        const float* __restrict__ W4, const float* __restrict__ B4,
                        float* __restrict__ OUT, int N)
{
    const int lane = threadIdx.x;      // 0..31 (wave32)
    const int j0   = blockIdx.x * 16;  // output col tile
    const int i0   = blockIdx.y * 16;  // output row tile
    const int bb   = blockIdx.z;       // batch

    // ---- strictly-upper tile: output is identically zero --------------------
    if (j0 > i0) {
        int r = lane >> 1, cb = (lane & 1) * 8;      // 32 lanes cover 16x16
        size_t base = ((size_t)bb * N + (i0 + r)) * N + j0 + cb;
        float4 z = {0.f, 0.f, 0.f, 0.f};
        *(float4*)(OUT + base)     = z;
        *(float4*)(OUT + base + 4) = z;
        return;
    }
    const bool diagTile = (i0 == j0);                // uniform across the wave

    // LDS staging: 18x18 halo of h1 (8ch) + 17x17 of h2 (16ch) + masks. ~30 KB.
    __shared__ float sH1[18][18][8];
    __shared__ float sH2[17][17][16];
    __shared__ unsigned char sM [18][18];
    __shared__ unsigned char sM2[17][17];
    __shared__ unsigned char sM3[16][16];

    const int lh = lane >> 4;          // lane half: selects K pair in A/B frags
    const int ln = lane & 15;          // M-site (A) / N-channel (B,D) index

    const float a1 = A1[0], a2 = A2[0], a3 = A3[0];
    const float b4 = B4[0];

    // ---- weight fragments in registers (B matrices, KxN layout) -------------
    // layer2: k = tap*8 + cin  (tap = di*2+dj), N = 16 out-ch
    v2f bw2[8];
#pragma unroll
    for (int t = 0; t < 8; ++t) {
        int ks = 4 * t + 2 * lh;
        int c0 = ks & 7,        tp0 = ks >> 3;
        int c1 = (ks + 1) & 7,  tp1 = (ks + 1) >> 3;
        bw2[t].x = W2[((ln * 8 + c0) * 2 + (tp0 >> 1)) * 2 + (tp0 & 1)];
        bw2[t].y = W2[((ln * 8 + c1) * 2 + (tp1 >> 1)) * 2 + (tp1 & 1)];
    }
    // layer3: k = tap*16 + cin, N padded 8 -> 16 with zeros
    v2f bw3[16];
#pragma unroll
    for (int t = 0; t < 16; ++t) {
        int ks = 4 * t + 2 * lh;
        int c0 = ks & 15,       tp0 = ks >> 4;
        int c1 = (ks + 1) & 15, tp1 = (ks + 1) >> 4;
        float wx = 0.f, wy = 0.f;
        if (ln < 8) {
            wx = W3[((ln * 16 + c0) * 2 + (tp0 >> 1)) * 2 + (tp0 & 1)];
            wy = W3[((ln * 16 + c1) * 2 + (tp1 >> 1)) * 2 + (tp1 & 1)];
        }
        bw3[t].x = wx; bw3[t].y = wy;
    }
    const float b2ln = B2[ln];
    const float b3ln = (ln < 8) ? B3[ln] : 0.f;
    const float w4ln = (ln < 8) ? W4[ln] : 0.f;

    // ---- phase 1: h1 = prelu(a1, (w1*x*m + b1) * m) over 18x18 halo ---------
    for (int s = lane; s < 18 * 18; s += 32) {
        int rr = s / 18, cc = s % 18;
        int gr = i0 - 1 + rr, gc = j0 - 1 + cc;
        float xv = 0.f; int mv = 0;
        if (gr >= 0 && gr < N && gc >= 0 && gc < N) {
            size_t gi = ((size_t)bb * N + gr) * N + gc;
            xv = X[gi];
            mv = (MASK[gi] > 0) ? 1 : 0;
        }
        sM[rr][cc] = (unsigned char)mv;
        float m = (float)mv;
        float xm = xv * m;
#pragma unroll
        for (int c = 0; c < 8; ++c) {
            float t = (W1[c] * xm + B1[c]) * m;
            sH1[rr][cc][c] = (t >= 0.f) ? t : a1 * t;
        }
    }
    __syncthreads();

    // ---- phase 2: m2 spread mask on 17x17 h2 region -------------------------
    for (int s = lane; s < 17 * 17; s += 32) {
        int r = s / 17, c = s % 17;
        int gr2 = i0 + r, gc2 = j0 - 1 + c;                  // global h2 coords
        int ok = (gr2 <= N) && (gc2 >= 0) && (gc2 <= N - 2); // h2 domain [N+1,N-1]
        int any = sM[r][c] | sM[r][c + 1] | sM[r + 1][c] | sM[r + 1][c + 1];
        sM2[r][c] = (unsigned char)((ok && any) ? 1 : 0);
    }
    __syncthreads();

    // ---- phase 3: m3 spread mask on 16x16 output tile -----------------------
    for (int s = lane; s < 256; s += 32) {
        int g = s >> 4, jc = s & 15;
        sM3[g][jc] = (unsigned char)((sM2[g][jc] | sM2[g][jc + 1] |
                                      sM2[g + 1][jc] | sM2[g + 1][jc + 1]) ? 1 : 0);
    }
    __syncthreads();

    // ---- phase 4: layer2 (K=32) via v_wmma_f32_16x16x4_f32 ------------------
    // 289 h2 sites packed 16/WMMA-group; M-row = arbitrary site.
    for (int g = 0; g < 19; ++g) {
        int idx = g * 16 + ln; idx = (idx < 288) ? idx : 288;  // clamp (dup safe)
        int r = idx / 17, c = idx % 17;                        // h2 region coords
        v8f acc = {0.f, 0.f, 0.f, 0.f, 0.f, 0.f, 0.f, 0.f};
#pragma unroll
        for (int t = 0; t < 8; ++t) {
            int ks = 4 * t + 2 * lh;
            int tap = ks >> 3, ch = ks & 7;
            int di = tap >> 1, dj = tap & 1;
            v2f a = *(const v2f*)&sH1[r + di][c + dj][ch];     // ds_load_b64
            acc = __builtin_amdgcn_wmma_f32_16x16x4_f32(
                      false, a, false, bw2[t], (short)0, acc, false, false);
        }
        // epilogue: D elem v in lane -> site (g*16 + v + 8*lh), channel ln
#pragma unroll
        for (int v = 0; v < 8; ++v) {
            int sidx = g * 16 + v + 8 * lh;
            sidx = (sidx < 288) ? sidx : 288;                  // dup writes identical
            int rr = sidx / 17, cc = sidx % 17;
            float m2v = (float)sM2[rr][cc];
            float t = (acc[v] + b2ln) * m2v;
            sH2[rr][cc][ln] = (t >= 0.f) ? t : a2 * t;
        }
    }
    __syncthreads();

    // ---- phase 5: layer3 (K=64) WMMA + fused 1x1 conv4 / mask / softplus ----
    for (int g = 0; g < 16; ++g) {                 // output row i = i0+g
        v8f acc = {0.f, 0.f, 0.f, 0.f, 0.f, 0.f, 0.f, 0.f};
#pragma unroll
        for (int t = 0; t < 16; ++t) {
            int ks = 4 * t + 2 * lh;
            int tap = ks >> 4, ch = ks & 15;
            int di = tap >> 1, dj = tap & 1;
            // site column jc = ln; h2 region coords (g+di, jc+dj)
            v2f a = *(const v2f*)&sH2[g + di][ln + dj][ch];
            acc = __builtin_amdgcn_wmma_f32_16x16x4_f32(
                      false, a, false, bw3[t], (short)0, acc, false, false);
        }
        float sums[8];
#pragma unroll
        for (int v = 0; v < 8; ++v) {
            int jc2 = v + 8 * lh;                  // column within tile
            float m3v = (float)sM3[g][jc2];
            float t = (acc[v] + b3ln) * m3v;       // bias, mask
            t = (t >= 0.f) ? t : a3 * t;           // prelu a3
            float contrib = w4ln * t;              // layer4 1x1 conv partial
            // butterfly reduce over the 16 channels striped across the half
            contrib += __shfl_xor(contrib, 1, 32);
            contrib += __shfl_xor(contrib, 2, 32);
            contrib += __shfl_xor(contrib, 4, 32);
            contrib += __shfl_xor(contrib, 8, 32);
            float o = (contrib + b4) * m3v;
            if (diagTile) {                        // uniform branch: 64 tiles only
                int i = i0 + g, j = j0 + jc2;
                o = (i >= j) ? o : 0.f;            // lower-triangular keep
                float sp = log1pf(expf(-fabsf(o))) + fmaxf(o, 0.f); // stable softplus
                o = (i == j && m3v > 0.5f) ? sp : o;
            }
            sums[v] = o;
        }
        if (ln == 0) {                             // lanes 0 & 16 write 8 cols each
            size_t base = ((size_t)bb * N + (i0 + g)) * N + j0 + 8 * lh;
            float4 o0 = {sums[0], sums[1], sums[2], sums[3]};
            float4 o1 = {sums[4], sums[5], sums[6], sums[7]};
            *(float4*)(OUT + base)     = o0;
            *(float4*)(OUT + base + 4) = o1;
        }
    }
}

extern "C" void kernel_launch(void* const* d_in, const int* in_sizes, int n_in,
                              void* d_out, int out_size, void* d_ws, size_t ws_size,
                              hipStream_t stream) {
    const float* X  = (const float*)d_in[0];
    const int*   Mk = (const int*)  d_in[1];
    const float* W1 = (const float*)d_in[2];
    const float* B1 = (const float*)d_in[3];
    const float* A1 = (const float*)d_in[4];
    const float* W2 = (const float*)d_in[5];
    const float* B2 = (const float*)d_in[6];
    const float* A2 = (const float*)d_in[7];
    const float* W3 = (const float*)d_in[8];
    const float* B3 = (const float*)d_in[9];
    const float* A3 = (const float*)d_in[10];
    const float* W4 = (const float*)d_in[11];
    const float* B4 = (const float*)d_in[12];
    float* OUT = (float*)d_out;

    const int N = 1024;
    const int B = in_sizes[0] / (N * N);           // channel dim is 1
    dim3 grid(N / 16, N / 16, B);
    precond_net_kernel<<<grid, 32, 0, stream>>>(X, Mk, W1, B1, A1, W2, B2, A2,
                                                W3, B3, A3, W4, B4, OUT, N);
}